// FoundationObservationEmbedding_28518582845510
// MI455X (gfx1250) — compile-verified
//
#include <hip/hip_runtime.h>
#include <hip/hip_bf16.h>
#include <stdint.h>

// FoundationObservationEmbedding for MI455X (gfx1250, wave32).
// Pipeline: threefry2x32 RNG -> bucket sort (exact top_k order incl. ties)
// -> embedding row gather via CDNA5 async LDS copies (ASYNCcnt path).

#define VOCAB  65536u
#define BATCH  64u
#define DIM    256u
#define NOBS   8192u
#define NBUCK  2048u            // buckets on high 11 bits of inverted 23-bit key
#define CAP    96u              // mean occupancy 32; P(overflow) ~ e^-43 per bucket
#define NTOT   (BATCH * VOCAB)  // 4,194,304 uniform draws
#define HALF   (NTOT / 2u)      // threefry pairs (JAX splits iota in half)

// ---------------------------------------------------------------- threefry
__device__ __forceinline__ unsigned rotl32(unsigned x, unsigned n) {
  return (x << n) | (x >> (32u - n));
}

// Exact JAX threefry2x32-20 (key = threefry_seed(42) = {0, 42}).
__device__ __forceinline__ void threefry2x32(unsigned k0, unsigned k1,
                                             unsigned x0, unsigned x1,
                                             unsigned& o0, unsigned& o1) {
  const unsigned ks2 = k0 ^ k1 ^ 0x1BD11BDAu;
  x0 += k0; x1 += k1;
#define TF4(a,b,c,d) \
  x0 += x1; x1 = rotl32(x1,(a)); x1 ^= x0; \
  x0 += x1; x1 = rotl32(x1,(b)); x1 ^= x0; \
  x0 += x1; x1 = rotl32(x1,(c)); x1 ^= x0; \
  x0 += x1; x1 = rotl32(x1,(d)); x1 ^= x0;
  TF4(13,15,26,6)   x0 += k1;  x1 += ks2 + 1u;
  TF4(17,29,16,24)  x0 += ks2; x1 += k0  + 2u;
  TF4(13,15,26,6)   x0 += k0;  x1 += k1  + 3u;
  TF4(17,29,16,24)  x0 += k1;  x1 += ks2 + 4u;
  TF4(13,15,26,6)   x0 += ks2; x1 += k0  + 5u;
#undef TF4
  o0 = x0; o1 = x1;
}

// u = bitcast(0x3f800000 | (bits>>9)) - 1 is monotonic in r = bits>>9, so
// top_k order == sort by r desc, index asc on ties (top_k is stable).
// We sort ascending by inv = 0x7FFFFF - r with index as low-order tiebreak.
__device__ __forceinline__ void bucket_insert(unsigned flat, unsigned bits,
                                              unsigned* cnt, unsigned* slots) {
  const unsigned r    = bits >> 9;
  const unsigned inv  = 0x7FFFFFu - r;
  const unsigned b    = flat >> 16;          // row (VOCAB == 2^16)
  const unsigned v    = flat & 0xFFFFu;      // vocab index
  const unsigned bk   = inv >> 12;           // 11-bit bucket id
  const unsigned key  = ((inv & 0xFFFu) << 16) | v;  // 12 low key bits + index
  const unsigned cell = b * NBUCK + bk;
  const unsigned pos  = atomicAdd(&cnt[cell], 1u);
  if (pos < CAP) slots[cell * CAP + pos] = key;
}

// -------------------------------------------------- kernel 1: RNG + bucketize
__global__ __launch_bounds__(256) void k_rng_bucket(unsigned* __restrict__ cnt,
                                                    unsigned* __restrict__ slots) {
  const unsigned j = blockIdx.x * 256u + threadIdx.x;   // grid == HALF threads
  unsigned o0, o1;
  threefry2x32(0u, 42u, j, j + HALF, o0, o1);           // counts = iota split in half
  bucket_insert(j,        o0, cnt, slots);
  bucket_insert(j + HALF, o1, cnt, slots);
}

// ---------------------- kernel 2: per-row prefix sum + bucket sort + emit idx
__global__ __launch_bounds__(1024) void k_sort_emit(const unsigned* __restrict__ cnt,
                                                    const unsigned* __restrict__ slots,
                                                    unsigned* __restrict__ sortedIdx) {
  const unsigned b = blockIdx.x;
  const unsigned t = threadIdx.x;
  __shared__ unsigned sc[NBUCK];     // clipped counts
  __shared__ unsigned soff[NBUCK];   // exclusive offsets
  __shared__ unsigned sp[1024];      // pair-sum scan

  for (unsigned k = t; k < NBUCK; k += 1024u) {
    unsigned c = cnt[b * NBUCK + k];
    sc[k] = c < CAP ? c : CAP;
  }
  __syncthreads();

  const unsigned c0 = sc[2u * t], c1 = sc[2u * t + 1u];
  sp[t] = c0 + c1;
  __syncthreads();
  for (unsigned d = 1u; d < 1024u; d <<= 1) {       // Hillis-Steele inclusive scan
    const unsigned add = (t >= d) ? sp[t - d] : 0u;
    __syncthreads();
    sp[t] += add;
    __syncthreads();
  }
  const unsigned excl = sp[t] - (c0 + c1);
  soff[2u * t]      = excl;
  soff[2u * t + 1u] = excl + c0;
  __syncthreads();

  // Only buckets whose range intersects [0, NOBS) need sorting (~first 256).
  for (unsigned k = t; k < NBUCK; k += 1024u) {
    const unsigned base = soff[k];
    const unsigned n    = sc[k];
    if (base >= NOBS || n == 0u) continue;
    unsigned a[CAP];
    const unsigned* sl = &slots[((size_t)b * NBUCK + k) * CAP];
    for (unsigned i = 0; i < n; ++i) a[i] = sl[i];
    for (unsigned i = 1; i < n; ++i) {              // insertion sort, ~32 elems
      const unsigned key = a[i];
      int m = (int)i - 1;
      while (m >= 0 && a[m] > key) { a[m + 1] = a[m]; --m; }
      a[m + 1] = key;
    }
    unsigned lim = NOBS - base; if (lim > n) lim = n;
    unsigned* dst = &sortedIdx[(size_t)b * NOBS + base];
    for (unsigned i = 0; i < lim; ++i) dst[i] = a[i] & 0xFFFFu;
  }
}

// ------------------------------- kernel 3: gather via CDNA5 async LDS copies
#define ASYNC_GATHER 1

__global__ __launch_bounds__(256) void k_gather(const float* __restrict__ emb,
                                                const unsigned* __restrict__ sortedIdx,
                                                float* __restrict__ out) {
  __shared__ float buf[8][DIM];                    // 1 KB per wave, 8 waves/block
  const unsigned wave = threadIdx.x >> 5;
  const unsigned lane = threadIdx.x & 31u;
  const unsigned row  = blockIdx.x * 8u + wave;    // 524,288 rows total (exact)
  const unsigned idx  = sortedIdx[row];            // wave-uniform

  const unsigned long long src =
      (unsigned long long)(const void*)(emb + (size_t)idx * DIM);
  const unsigned long long dst =
      (unsigned long long)(void*)(out + (size_t)row * DIM);

#if ASYNC_GATHER
  // Flat shared pointer low 32 bits == wave-relative LDS byte offset.
  const unsigned lds0  = (unsigned)(unsigned long long)(const void*)&buf[wave][0];
  const unsigned loff0 = lds0 + lane * 16u;
  const unsigned loff1 = loff0 + 512u;
  const unsigned long long g0 = src + lane * 16u;
  const unsigned long long g1 = g0 + 512u;
  // 32 lanes x 16B x 2 = 1 KB row  (ASYNCcnt-tracked, no VGPR staging)
  asm volatile("global_load_async_to_lds_b128 %0, %1, off"
               :: "v"(loff0), "v"(g0) : "memory");
  asm volatile("global_load_async_to_lds_b128 %0, %1, off"
               :: "v"(loff1), "v"(g1) : "memory");
  asm volatile("s_wait_asynccnt 0" ::: "memory");  // LDS writes visible
  const unsigned long long d0 = dst + lane * 16u;
  const unsigned long long d1 = d0 + 512u;
  asm volatile("global_store_async_from_lds_b128 %0, %1, off"
               :: "v"(d0), "v"(loff0) : "memory");
  asm volatile("global_store_async_from_lds_b128 %0, %1, off"
               :: "v"(d1), "v"(loff1) : "memory");
  asm volatile("s_wait_asynccnt 0" ::: "memory");  // drain before endpgm
#else
  const float4* s4 = (const float4*)src;
  float4*       d4 = (float4*)dst;
  d4[lane]       = s4[lane];
  d4[lane + 32u] = s4[lane + 32u];
#endif
}

// ---------------------------------------------------------------- launcher
extern "C" void kernel_launch(void* const* d_in, const int* in_sizes, int n_in,
                              void* d_out, int out_size, void* d_ws, size_t ws_size,
                              hipStream_t stream) {
  (void)in_sizes; (void)n_in; (void)out_size; (void)ws_size;
  const float* emb = (const float*)d_in[0];
  float* out = (float*)d_out;

  uint8_t* ws = (uint8_t*)d_ws;
  const size_t cnt_bytes   = (size_t)BATCH * NBUCK * sizeof(unsigned);        // 512 KB
  const size_t slots_bytes = (size_t)BATCH * NBUCK * CAP * sizeof(unsigned);  // 48 MB
  unsigned* cnt   = (unsigned*)(ws);
  unsigned* slots = (unsigned*)(ws + cnt_bytes);
  unsigned* sidx  = (unsigned*)(ws + cnt_bytes + slots_bytes);                // 2 MB

  hipMemsetAsync(cnt, 0, cnt_bytes, stream);        // deterministic each call
  k_rng_bucket<<<HALF / 256u, 256, 0, stream>>>(cnt, slots);
  k_sort_emit<<<BATCH, 1024, 0, stream>>>(cnt, slots, sidx);
  k_gather<<<(BATCH * NOBS) / 8u, 256, 0, stream>>>(emb, sidx, out);
}